// gaussianMixtureModel_76708115907079
// MI455X (gfx1250) — compile-verified
//
#include <hip/hip_runtime.h>

#define K_CL   512
#define DIMS   32
#define ROWS   16
#define LOG_2PI 1.8378770664093453f

typedef __attribute__((ext_vector_type(2))) float v2f;
typedef __attribute__((ext_vector_type(8))) float v8f;

__device__ __forceinline__ float sigmoidf_(float x) { return 1.0f / (1.0f + expf(-x)); }

// ---------------------------------------------------------------------------
// P1: per-cluster precompute. Logical W (64xK): rows 0..31 = -0.5/var (coeff
// of z^2), rows 32..63 = mu/var (coeff of z). Stored PACKED in WMMA-fragment
// order: Wpack[((t*8 + p)*32 + lane)*4 + pos], where tile t = k/16, pair p
// covers logical rows 8p..8p+7, lane = h*16 + (k%16), pos selects the two
// fragments {8p+2h, 8p+2h+1, 8p+4+2h, 8p+4+2h+1}. One b128 per lane per pair.
// C[k] = D*logp - 0.5*sum(log2pi+logvar) - 0.5*sum mu^2/var. Also log_prior.
// ---------------------------------------------------------------------------
__device__ __forceinline__ void scatter_pack(float* __restrict__ Wpack,
                                             int r, int k, float val) {
    int t = k >> 4;
    int n = k & 15;
    int p   = r >> 3;
    int rem = r & 7;
    int s   = (rem >> 2) & 1;     // which fragment of the pair
    int h   = (rem >> 1) & 1;     // lane half
    int q   = rem & 1;            // which of the 2 floats in the fragment
    int lane = h * 16 + n;
    Wpack[(((t * 8 + p) * 32) + lane) * 4 + (s * 2 + q)] = val;
}

__global__ void gmm_precompute_cols(const float* __restrict__ cmean,
                                    const float* __restrict__ cvsu,
                                    const float* __restrict__ cprior,
                                    float* __restrict__ Wpack,
                                    float* __restrict__ Ck,
                                    float* __restrict__ logp) {
    int k = blockIdx.x * blockDim.x + threadIdx.x;
    if (k >= K_CL) return;
    float pr = sigmoidf_(cprior[k]);
    float lp = logf(pr);
    logp[k] = lp;
    float c = (float)DIMS * lp;
    for (int d = 0; d < DIMS; ++d) {
        float v  = sigmoidf_(cvsu[d * K_CL + k]);
        float lv = logf(v);
        float iv = 1.0f / v;
        float mu = cmean[d * K_CL + k];
        c -= 0.5f * (LOG_2PI + lv);
        c -= 0.5f * mu * mu * iv;
        scatter_pack(Wpack, d,        k, -0.5f * iv);   // coeff of z^2
        scatter_pack(Wpack, DIMS + d, k,  mu * iv);     // coeff of z
    }
    Ck[k] = c;
}

// ---------------------------------------------------------------------------
// P2: per-dim sums over k: s1[d]=sum 1/var, s2[d]=sum mu/var, plus scalar
// K2 = 0.5*D^2*log2pi + sum(logvar) + sum(mu^2/var).
// ---------------------------------------------------------------------------
__global__ void gmm_precompute_dims(const float* __restrict__ cmean,
                                    const float* __restrict__ cvsu,
                                    float* __restrict__ s1a,
                                    float* __restrict__ s2a,
                                    float* __restrict__ k2out) {
    __shared__ float sh3[DIMS], shl[DIMS];
    int d = threadIdx.x;
    float s1 = 0.f, s2 = 0.f, s3 = 0.f, slv = 0.f;
    for (int k = 0; k < K_CL; ++k) {
        float v  = sigmoidf_(cvsu[d * K_CL + k]);
        float iv = 1.0f / v;
        float mu = cmean[d * K_CL + k];
        s1 += iv; s2 += mu * iv; s3 += mu * mu * iv; slv += logf(v);
    }
    s1a[d] = s1;
    s2a[d] = s2;
    sh3[d] = s3;
    shl[d] = slv;
    __syncthreads();
    if (d == 0) {
        float t3 = 0.f, tl = 0.f;
        for (int i = 0; i < DIMS; ++i) { t3 += sh3[i]; tl += shl[i]; }
        k2out[0] = 0.5f * (float)(DIMS * DIMS) * LOG_2PI + tl + t3;
    }
}

// ---------------------------------------------------------------------------
// Main: 1 block = 16 rows x 512 clusters. 8 waves; wave w owns 4 N-tiles.
// S tile via 16 chained V_WMMA_F32_16X16X4_F32 fed by 8 coalesced b128 loads.
// ---------------------------------------------------------------------------
__global__ __launch_bounds__(256) void gmm_main(
    const float* __restrict__ z_mean,
    const float* __restrict__ zlv,
    const float* __restrict__ z,
    const float* __restrict__ Wpack,
    const float* __restrict__ Ck,
    const float* __restrict__ logp,
    const float* __restrict__ s1a,
    const float* __restrict__ s2a,
    const float* __restrict__ k2p,
    float* __restrict__ P_out,
    float* __restrict__ nloss) {
    __shared__ __align__(16) float U_lds[ROWS * K_CL];  // [m][k] exp(S)+eps
    __shared__ float A_lds[64 * 16];                    // [k][m]
    __shared__ float zm_lds[ROWS * DIMS];
    __shared__ float zl_lds[ROWS * DIMS];
    __shared__ float rowU[ROWS], rowUS[ROWS], rowULP[ROWS], invZ[ROWS];

    const int tid = threadIdx.x;
    const int b0  = blockIdx.x * ROWS;

    if (tid < ROWS) { rowU[tid] = 0.f; rowUS[tid] = 0.f; rowULP[tid] = 0.f; }

    // Cache z_mean / z_log_var rows; build A (16 rows x 64 cols):
    // cols 0..31 -> z^2, cols 32..63 -> z.
    for (int e = tid; e < ROWS * DIMS; e += 256) {
        zm_lds[e] = z_mean[b0 * DIMS + e];
        zl_lds[e] = zlv[b0 * DIMS + e];
    }
    for (int e = tid; e < 64 * 16; e += 256) {
        int kr = e >> 4;        // 0..63
        int m  = e & 15;
        float zz = z[(b0 + m) * DIMS + (kr & (DIMS - 1))];
        A_lds[e] = (kr < DIMS) ? zz * zz : zz;
    }
    __syncthreads();

    const int lane = tid & 31;
    const int wave = tid >> 5;
    const int n    = lane & 15;   // also A's m index
    const int half = lane >> 4;

    // Preload all 16 A fragments (16x4 f32 A layout: v0/v1 = K{0,1} for lanes
    // 0-15, K{2,3} for lanes 16-31).
    v2f afrag[16];
#pragma unroll
    for (int kk = 0; kk < 16; ++kk) {
        int kr = 4 * kk + 2 * half;
        afrag[kk].x = A_lds[kr * 16 + n];
        afrag[kk].y = A_lds[(kr + 1) * 16 + n];
    }

    float sU[8], sUS[8], sULP[8];
#pragma unroll
    for (int i = 0; i < 8; ++i) { sU[i] = 0.f; sUS[i] = 0.f; sULP[i] = 0.f; }

    for (int j = 0; j < 4; ++j) {
        const int t  = wave * 4 + j;
        const int n0 = t * 16;

        // 8 coalesced b128 loads cover the whole 64x16 B tile.
        const float4* wp = (const float4*)Wpack + (size_t)t * 8 * 32 + lane;
        float4 w[8];
#pragma unroll
        for (int p = 0; p < 8; ++p) w[p] = wp[p * 32];

        v8f acc = {0.f, 0.f, 0.f, 0.f, 0.f, 0.f, 0.f, 0.f};
#pragma unroll
        for (int p = 0; p < 8; ++p) {
            v2f bA; bA.x = w[p].x; bA.y = w[p].y;
            v2f bB; bB.x = w[p].z; bB.y = w[p].w;
            acc = __builtin_amdgcn_wmma_f32_16x16x4_f32(
                false, afrag[2 * p],     false, bA, (short)0, acc, false, false);
            acc = __builtin_amdgcn_wmma_f32_16x16x4_f32(
                false, afrag[2 * p + 1], false, bB, (short)0, acc, false, false);
        }

        const int ncol = n0 + n;
        const float cc = Ck[ncol];
        const float lp = logp[ncol];
#pragma unroll
        for (int i = 0; i < 8; ++i) {
            float s = acc[i] + cc;               // terms.sum(axis=1)
            float u = __expf(s) + 1e-10f;        // P_unnorm
            sU[i]   += u;
            sUS[i]  += u * s;                    // for sum P*logP
            sULP[i] += u * lp;                   // for sum P*log_prior
            U_lds[(i + 8 * half) * K_CL + ncol] = u;
        }
    }

#pragma unroll
    for (int i = 0; i < 8; ++i) {
        int row = i + 8 * half;
        atomicAdd(&rowU[row],   sU[i]);
        atomicAdd(&rowUS[row],  sUS[i]);
        atomicAdd(&rowULP[row], sULP[i]);
    }
    __syncthreads();

    if (tid < ROWS) {
        const int row = tid;
        float Z  = rowU[row];
        float iZ = 1.0f / Z;
        invZ[row] = iZ;
        float logZ  = logf(Z);
        float third = (rowUS[row] - rowULP[row]) * iZ - logZ;
        float second = k2p[0];
        float forth  = 0.f;
        const int lb = row * DIMS;
        for (int d = 0; d < DIMS; ++d) {
            float lv_ = zl_lds[lb + d];
            float zm  = zm_lds[lb + d];
            second += (__expf(lv_) + zm * zm) * s1a[d] - 2.0f * zm * s2a[d];
            forth  += 0.5f * (lv_ + 1.0f);
        }
        // negative_loss = second - third - forth
        nloss[b0 + row] = second - third - forth;
    }
    __syncthreads();

    // Normalize and store P as b128, coalesced over k.
    for (int e = tid; e < ROWS * K_CL / 4; e += 256) {
        int m = e >> 7;                         // 128 float4 per row
        float4 u = ((const float4*)U_lds)[e];
        float iz = invZ[m];
        float4 o; o.x = u.x * iz; o.y = u.y * iz; o.z = u.z * iz; o.w = u.w * iz;
        ((float4*)P_out)[(size_t)(b0 + m) * (K_CL / 4) + (e & 127)] = o;
    }
}

extern "C" void kernel_launch(void* const* d_in, const int* in_sizes, int n_in,
                              void* d_out, int out_size, void* d_ws, size_t ws_size,
                              hipStream_t stream) {
    const float* z_mean = (const float*)d_in[0];
    const float* zlv    = (const float*)d_in[1];
    const float* z      = (const float*)d_in[2];
    const float* cmean  = (const float*)d_in[3];
    const float* cvsu   = (const float*)d_in[4];
    const float* cprior = (const float*)d_in[5];

    const int B = in_sizes[0] / DIMS;    // 8192

    float* ws    = (float*)d_ws;
    float* Wpack = ws;                   // 64*512 packed
    float* CkP   = Wpack + 64 * K_CL;    // 512
    float* logp  = CkP + K_CL;           // 512
    float* s1a   = logp + K_CL;          // 32
    float* s2a   = s1a + DIMS;           // 32
    float* k2p   = s2a + DIMS;           // 1

    float* P_out = (float*)d_out;                // (B, K)
    float* nloss = P_out + (size_t)B * K_CL;     // (B,)

    gmm_precompute_cols<<<(K_CL + 255) / 256, 256, 0, stream>>>(
        cmean, cvsu, cprior, Wpack, CkP, logp);
    gmm_precompute_dims<<<1, DIMS, 0, stream>>>(cmean, cvsu, s1a, s2a, k2p);
    gmm_main<<<B / ROWS, 256, 0, stream>>>(
        z_mean, zlv, z, Wpack, CkP, logp, s1a, s2a, k2p, P_out, nloss);
}